// AKT_56349970924168
// MI455X (gfx1250) — compile-verified
//
#include <hip/hip_runtime.h>
#include <math.h>

typedef float v2f __attribute__((ext_vector_type(2)));
typedef float v8f __attribute__((ext_vector_type(8)));

#define BB   2
#define SSQ  2048
#define DDIM 512
#define HHH  8
#define DKK  64
#define DFFN 2048
#define MMR  (BB * SSQ)   // 4096 rows

// ---------------------------------------------------------------------------
// WMMA fp32 GEMM:  C[M,N] = A[M,K] @ W[K,N] + bias[N]   (optional ReLU)
// Block tile 128x64, 8 waves in 4x2; each wave owns a 32x32 strip via a
// 2x2 grid of f32 16x16x4 accumulators (4 WMMAs / 8 LDS dwords per k-step).
// K staged through LDS in 32-wide chunks with float4 (b128) copies.
// M multiple of 128, N multiple of 64, K multiple of 32 (true for all calls).
// ---------------------------------------------------------------------------
__global__ __launch_bounds__(256)
void akt_gemm_wmma(const float* __restrict__ A, const float* __restrict__ W,
                   const float* __restrict__ bias, float* __restrict__ C,
                   int M, int N, int K, int relu)
{
    __shared__ float As[128][36];   // row stride 144B: 16B aligned, conflict-free frag reads
    __shared__ float Bs[32][68];    // row stride 272B: 16B aligned

    const int tid  = threadIdx.x;
    const int lane = tid & 31;
    const int wave = tid >> 5;
    const int m0   = blockIdx.y * 128;
    const int n0   = blockIdx.x * 64;
    const int wm   = (wave & 3) * 32;   // wave row offset in tile
    const int wn   = (wave >> 2) * 32;  // wave col offset in tile
    const int lr   = lane & 15;
    const int kp   = (lane >> 4) * 2;   // lanes 0-15 -> K0,K1 ; lanes 16-31 -> K2,K3

    // ---- precompute staging pointers (advance by 32-col chunk each iter) ----
    // A tile: 128x32 floats = 1024 float4, 4 per thread
    const float4* aSrc[4];  float* aDst[4];
    #pragma unroll
    for (int j = 0; j < 4; ++j) {
        int e  = tid + 256 * j;
        int r  = e >> 3;            // 8 float4 per 32-col row
        int c4 = e & 7;
        aSrc[j] = (const float4*)(A + (size_t)(m0 + r) * K + c4 * 4);
        aDst[j] = &As[r][c4 * 4];
    }
    // B tile: 32x64 floats = 512 float4, 2 per thread
    const float4* bSrc[2];  float* bDst[2];
    #pragma unroll
    for (int j = 0; j < 2; ++j) {
        int e  = tid + 256 * j;
        int r  = e >> 4;            // 16 float4 per 64-col row
        int c4 = e & 15;
        bSrc[j] = (const float4*)(W + (size_t)r * N + n0 + c4 * 4);
        bDst[j] = &Bs[r][c4 * 4];
    }
    const int aAdv = 8;                 // 32 floats = 8 float4 along K
    const int bAdv = 8 * N;             // 32 rows * N floats = 8N float4

    v8f acc00 = {}, acc01 = {}, acc10 = {}, acc11 = {};

    for (int k0 = 0; k0 < K; k0 += 32) {
        #pragma unroll
        for (int j = 0; j < 4; ++j) { *(float4*)aDst[j] = *aSrc[j]; aSrc[j] += aAdv; }
        #pragma unroll
        for (int j = 0; j < 2; ++j) { *(float4*)bDst[j] = *bSrc[j]; bSrc[j] += bAdv; }
        __syncthreads();

        #pragma unroll
        for (int kk = 0; kk < 32; kk += 4) {
            v2f a0, a1, b0, b1;
            a0.x = As[wm + lr][kk + kp];
            a0.y = As[wm + lr][kk + kp + 1];
            a1.x = As[wm + 16 + lr][kk + kp];
            a1.y = As[wm + 16 + lr][kk + kp + 1];
            b0.x = Bs[kk + kp][wn + lr];
            b0.y = Bs[kk + kp + 1][wn + lr];
            b1.x = Bs[kk + kp][wn + 16 + lr];
            b1.y = Bs[kk + kp + 1][wn + 16 + lr];
            acc00 = __builtin_amdgcn_wmma_f32_16x16x4_f32(false, a0, false, b0,
                                                          (short)0, acc00, false, false);
            acc01 = __builtin_amdgcn_wmma_f32_16x16x4_f32(false, a0, false, b1,
                                                          (short)0, acc01, false, false);
            acc10 = __builtin_amdgcn_wmma_f32_16x16x4_f32(false, a1, false, b0,
                                                          (short)0, acc10, false, false);
            acc11 = __builtin_amdgcn_wmma_f32_16x16x4_f32(false, a1, false, b1,
                                                          (short)0, acc11, false, false);
        }
        __syncthreads();
    }

    // D layout: VGPR i -> lanes 0-15: (M=i, N=lane) ; lanes 16-31: (M=8+i, N=lane-16)
    const int rbase = m0 + wm + ((lane >> 4) << 3);
    const int c0    = n0 + wn + lr;
    const float bia0 = bias[c0];
    const float bia1 = bias[c0 + 16];
    #pragma unroll
    for (int i = 0; i < 8; ++i) {
        int ra = rbase + i;
        int rb = ra + 16;
        float v00 = acc00[i] + bia0;
        float v01 = acc01[i] + bia1;
        float v10 = acc10[i] + bia0;
        float v11 = acc11[i] + bia1;
        if (relu) {
            v00 = fmaxf(v00, 0.f); v01 = fmaxf(v01, 0.f);
            v10 = fmaxf(v10, 0.f); v11 = fmaxf(v11, 0.f);
        }
        C[(size_t)ra * N + c0]      = v00;
        C[(size_t)ra * N + c0 + 16] = v01;
        C[(size_t)rb * N + c0]      = v10;
        C[(size_t)rb * N + c0 + 16] = v11;
    }
}

// ---------------------------------------------------------------------------
// block helpers (256 threads)
// ---------------------------------------------------------------------------
__device__ __forceinline__ float blk_max(float v, float* red, int tid) {
    red[tid] = v; __syncthreads();
    for (int off = 128; off > 0; off >>= 1) {
        if (tid < off) red[tid] = fmaxf(red[tid], red[tid + off]);
        __syncthreads();
    }
    float r = red[0]; __syncthreads(); return r;
}
__device__ __forceinline__ float blk_sum(float v, float* red, int tid) {
    red[tid] = v; __syncthreads();
    for (int off = 128; off > 0; off >>= 1) {
        if (tid < off) red[tid] += red[tid + off];
        __syncthreads();
    }
    float r = red[0]; __syncthreads(); return r;
}
// inclusive Hillis-Steele scan over 256 thread values; also returns grand total
__device__ __forceinline__ float blk_scan(float v, float* red, int tid, float* total) {
    red[tid] = v; __syncthreads();
    for (int off = 1; off < 256; off <<= 1) {
        float t = (tid >= off) ? red[tid - off] : 0.f;
        __syncthreads();
        red[tid] += t;
        __syncthreads();
    }
    float incl = red[tid];
    *total = red[255];
    __syncthreads();
    return incl;
}

// ---------------------------------------------------------------------------
// AKT attention, one block per (query-row, head, batch).
// qk: [B,S,D] (q == k, kq_same), v: [B,S,D]; out o: [B,S,D]
// ---------------------------------------------------------------------------
__global__ __launch_bounds__(256)
void akt_attn(const float* __restrict__ qk, const float* __restrict__ v,
              const float* __restrict__ gammas, int layer, int mask_k,
              float* __restrict__ o)
{
    __shared__ float sc[SSQ];     // probabilities row
    __shared__ float sq[DKK];     // query vector
    __shared__ float red[256];

    const int qi  = blockIdx.x;
    const int h   = blockIdx.y;
    const int b   = blockIdx.z;
    const int tid = threadIdx.x;

    const size_t base = ((size_t)b * SSQ) * DDIM + (size_t)h * DKK; // + row*DDIM
    float* orow = o + base + (size_t)qi * DDIM;

    if (mask_k == 0 && qi == 0) {           // zero_pad: first row output = 0
        if (tid < DKK) orow[tid] = 0.f;
        return;
    }
    const int kmax = (mask_k == 1) ? qi : qi - 1;   // inclusive causal bound

    if (tid < DKK) sq[tid] = qk[base + (size_t)qi * DDIM + tid];
    __syncthreads();

    // raw scores: each thread owns 8 contiguous k positions (float4 dot)
    const int k0 = tid * 8;
    const float4* sq4 = (const float4*)sq;
    float s_loc[8];
    #pragma unroll
    for (int j = 0; j < 8; ++j) {
        const float4* kr4 = (const float4*)(qk + base + (size_t)(k0 + j) * DDIM);
        float acc = 0.f;
        #pragma unroll
        for (int d4 = 0; d4 < DKK / 4; ++d4) {
            float4 qa = sq4[d4], kb = kr4[d4];
            acc += qa.x * kb.x + qa.y * kb.y + qa.z * kb.z + qa.w * kb.w;
        }
        s_loc[j] = acc * 0.125f;            // 1/sqrt(64)
    }

    // softmax #1 is over ALL k (reference masks after softmax)
    float m = s_loc[0];
    #pragma unroll
    for (int j = 1; j < 8; ++j) m = fmaxf(m, s_loc[j]);
    const float m1 = blk_max(m, red, tid);

    float sumAll = 0.f, incl = 0.f, inclArr[8];
    #pragma unroll
    for (int j = 0; j < 8; ++j) {
        float e = expf(s_loc[j] - m1);
        sumAll += e;
        float w = ((k0 + j) <= kmax) ? e : 0.f;   // scores_ = softmax * mask
        incl += w;
        inclArr[j] = incl;
    }
    const float sum1 = blk_sum(sumAll, red, tid);
    float totN;
    const float inclT  = blk_scan(incl, red, tid, &totN);
    const float prefix = inclT - incl;
    const float inv1 = 1.f / sum1;
    const float dtot = totN * inv1;               // disttotal

    // distance-decay modulation, masked scores for softmax #2
    const float gm = gammas[layer * HHH + h];
    const float g  = -log1pf(expf(gm));           // -softplus(gamma)
    float m2loc = -3.4e38f;
    float s2[8];
    #pragma unroll
    for (int j = 0; j < 8; ++j) {
        int k = k0 + j;
        if (k <= kmax) {
            float distcum = (prefix + inclArr[j]) * inv1;
            float rem  = dtot - distcum;
            float pos  = fabsf((float)(qi - k));
            float dist = sqrtf(fmaxf(rem * pos, 0.f));
            float eff  = expf(dist * g);
            eff = fminf(fmaxf(eff, 1e-5f), 1e5f);
            s2[j] = s_loc[j] * eff;
            m2loc = fmaxf(m2loc, s2[j]);
        } else {
            s2[j] = -3.4e38f;
        }
    }
    const float m2 = blk_max(m2loc, red, tid);

    float sum2loc = 0.f;
    #pragma unroll
    for (int j = 0; j < 8; ++j) {
        int k = k0 + j;
        float p = (k <= kmax) ? expf(s2[j] - m2) : 0.f;
        sc[k] = p;
        sum2loc += p;
    }
    const float sum2 = blk_sum(sum2loc, red, tid);  // syncs make sc visible
    const float inv2 = 1.f / sum2;

    // out[d] = sum_k p2[k] * V[k][d]  (4-way split over k, 64 d lanes coalesced)
    const int d   = tid & 63;
    const int grp = tid >> 6;
    const float* vb = v + base;
    float acc = 0.f;
    const int ks = grp * 512;
    for (int k = ks; k < ks + 512; ++k)
        acc += sc[k] * vb[(size_t)k * DDIM + d];
    red[tid] = acc;
    __syncthreads();
    if (tid < 64)
        orow[tid] = (red[tid] + red[tid + 64] + red[tid + 128] + red[tid + 192]) * inv2;
}

// ---------------------------------------------------------------------------
// out = LayerNorm(x + r) * s + b      (rows of width 512, one block per row)
// ---------------------------------------------------------------------------
__global__ __launch_bounds__(256)
void akt_add_ln(const float* __restrict__ x, const float* __restrict__ r,
                const float* __restrict__ s, const float* __restrict__ b,
                float* __restrict__ out)
{
    __shared__ float red[256];
    const int row = blockIdx.x;
    const int tid = threadIdx.x;
    const size_t o = (size_t)row * DDIM;

    float v0 = x[o + tid]       + r[o + tid];
    float v1 = x[o + tid + 256] + r[o + tid + 256];

    red[tid] = v0 + v1; __syncthreads();
    for (int off = 128; off > 0; off >>= 1) {
        if (tid < off) red[tid] += red[tid + off];
        __syncthreads();
    }
    const float mean = red[0] * (1.f / DDIM);
    __syncthreads();

    const float d0 = v0 - mean, d1 = v1 - mean;
    red[tid] = d0 * d0 + d1 * d1; __syncthreads();
    for (int off = 128; off > 0; off >>= 1) {
        if (tid < off) red[tid] += red[tid + off];
        __syncthreads();
    }
    const float inv = rsqrtf(red[0] * (1.f / DDIM) + 1e-5f);

    out[o + tid]       = d0 * inv * s[tid]       + b[tid];
    out[o + tid + 256] = d1 * inv * s[tid + 256] + b[tid + 256];
}

// ---------------------------------------------------------------------------
extern "C" void kernel_launch(void* const* d_in, const int* in_sizes, int n_in,
                              void* d_out, int out_size, void* d_ws, size_t ws_size,
                              hipStream_t stream)
{
    (void)in_sizes; (void)n_in; (void)out_size; (void)ws_size;

    const float* s_emb  = (const float*)d_in[0];
    const float* sa_emb = (const float*)d_in[1];
    const float* Wk     = (const float*)d_in[2];
    const float* bk     = (const float*)d_in[3];
    const float* Wv     = (const float*)d_in[4];
    const float* bv     = (const float*)d_in[5];
    const float* Wo     = (const float*)d_in[6];
    const float* bo     = (const float*)d_in[7];
    const float* gam    = (const float*)d_in[8];
    const float* ln1s   = (const float*)d_in[9];
    const float* ln1b   = (const float*)d_in[10];
    const float* W1     = (const float*)d_in[11];
    const float* b1     = (const float*)d_in[12];
    const float* W2     = (const float*)d_in[13];
    const float* b2     = (const float*)d_in[14];
    const float* ln2s   = (const float*)d_in[15];
    const float* ln2b   = (const float*)d_in[16];
    float* out          = (float*)d_out;

    const size_t BSD = (size_t)BB * SSQ * DDIM;      // 2,097,152
    float* ws   = (float*)d_ws;
    float* qbuf = ws;              // q (== k, kq_same)
    float* vbuf = ws + 1 * BSD;
    float* obuf = ws + 2 * BSD;
    float* pbuf = ws + 3 * BSD;
    float* t1   = ws + 4 * BSD;
    float* ybuf = ws + 5 * BSD;
    float* xbuf = ws + 6 * BSD;
    float* ffb  = ws + 7 * BSD;    // B*S*DFF floats

    auto gemm = [&](const float* A, const float* W, const float* bias,
                    float* C, int K, int N, int relu) {
        dim3 g(N / 64, MMR / 128);
        akt_gemm_wmma<<<g, 256, 0, stream>>>(A, W, bias, C, MMR, N, K, relu);
    };

    auto layer = [&](int i, int mask_k, const float* query, const float* values,
                     int apply_pos, float* dst) {
        gemm(query,  Wk + (size_t)i * DDIM * DDIM, bk + (size_t)i * DDIM, qbuf, DDIM, DDIM, 0);
        gemm(values, Wv + (size_t)i * DDIM * DDIM, bv + (size_t)i * DDIM, vbuf, DDIM, DDIM, 0);
        akt_attn<<<dim3(SSQ, HHH, BB), 256, 0, stream>>>(qbuf, vbuf, gam, i, mask_k, obuf);
        gemm(obuf, Wo + (size_t)i * DDIM * DDIM, bo + (size_t)i * DDIM, pbuf, DDIM, DDIM, 0);
        akt_add_ln<<<MMR, 256, 0, stream>>>(query, pbuf,
                                            ln1s + (size_t)i * DDIM, ln1b + (size_t)i * DDIM,
                                            apply_pos ? t1 : dst);
        if (apply_pos) {
            gemm(t1,  W1 + (size_t)i * DDIM * DFFN, b1 + (size_t)i * DFFN, ffb,  DDIM, DFFN, 1);
            gemm(ffb, W2 + (size_t)i * DFFN * DDIM, b2 + (size_t)i * DDIM, pbuf, DFFN, DDIM, 0);
            akt_add_ln<<<MMR, 256, 0, stream>>>(t1, pbuf,
                                                ln2s + (size_t)i * DDIM, ln2b + (size_t)i * DDIM,
                                                dst);
        }
    };

    layer(0, 1, sa_emb, sa_emb, 1, ybuf);   // blocks_1[0]
    layer(1, 1, s_emb,  s_emb,  0, xbuf);   // blocks_2[0]
    layer(2, 0, xbuf,   ybuf,   1, out);    // blocks_2[1]
}